// Encoding_83348135346610
// MI455X (gfx1250) — compile-verified
//
#include <hip/hip_runtime.h>

#define BATCH 16
#define CHAN  512
#define NPOS  4096   // 64*64
#define KCW   32

typedef float v2f __attribute__((ext_vector_type(2)));
typedef float v8f __attribute__((ext_vector_type(8)));

__device__ __forceinline__ v8f wmma_f32(v2f a, v2f b, v8f c) {
  // D = A(16x4 f32) * B(4x16 f32) + C(16x16 f32), full fp32 precision
  return __builtin_amdgcn_wmma_f32_16x16x4_f32(false, a, false, b, (short)0, c,
                                               false, false);
}

// ---------------------------------------------------------------- kernel 0
// c2[k] = sum_c cw[k][c]^2
__global__ __launch_bounds__(256) void enc_c2(const float* __restrict__ cw,
                                              float* __restrict__ c2) {
  __shared__ float part[8][KCW];
  const int t = threadIdx.x;
  const int k = t & (KCW - 1);
  const int p = t >> 5;
  float s = 0.f;
  for (int c = p * 64; c < (p + 1) * 64; ++c) {
    float v = cw[k * CHAN + c];
    s += v * v;
  }
  part[p][k] = s;
  __syncthreads();
  if (t < KCW) {
    float tot = 0.f;
#pragma unroll
    for (int i = 0; i < 8; ++i) tot += part[i][t];
    c2[t] = tot;
  }
}

// ---------------------------------------------------------------- kernel 1
// logits via f32 WMMA, softmax over K=32, write A[b][n][k]
__global__ __launch_bounds__(256) void enc_assign(
    const float* __restrict__ x, const float* __restrict__ cw,
    const float* __restrict__ scale, const float* __restrict__ c2,
    float* __restrict__ A) {
  __shared__ float lsm[8][16][33];               // per-wave 16n x 32k (padded)
  const int lane = threadIdx.x & 31;
  const int wave = threadIdx.x >> 5;
  const int tile = blockIdx.x * 8 + wave;        // B*N/16 tiles total
  const int b    = tile >> 8;                    // 256 tiles per batch
  const int n0   = (tile & 255) << 4;
  const int g    = lane >> 4;                    // half-wave id
  const int m    = lane & 15;
  const float* xb  = x + (size_t)b * CHAN * NPOS + n0 + m;
  const float* cw0 = cw + (size_t)m * CHAN + 2 * g;          // k-tile 0
  const float* cw1 = cw + (size_t)(m + 16) * CHAN + 2 * g;   // k-tile 1
  v8f acc0 = {};
  v8f acc1 = {};
  float x2p = 0.f;
  for (int c0 = 0; c0 < CHAN; c0 += 4) {
    // A-frag: element (row m, kdim j+2g) -> x[b][c0+2g+j][n0+m]
    float a0 = xb[(size_t)(c0 + 2 * g) * NPOS];
    float a1 = xb[(size_t)(c0 + 2 * g + 1) * NPOS];
    x2p += a0 * a0 + a1 * a1;                    // ||x||^2 partial (free ride)
    v2f af; af.x = a0; af.y = a1;
    // B-frags: element (kdim j+2g, col k=m) -> cw[k][c0+2g+j]
    v2f b0; b0.x = cw0[c0]; b0.y = cw0[c0 + 1];
    v2f b1; b1.x = cw1[c0]; b1.y = cw1[c0 + 1];
    acc0 = wmma_f32(af, b0, acc0);
    acc1 = wmma_f32(af, b1, acc1);
  }
  // stash xc tile: acc element (row r+8g, col m)
#pragma unroll
  for (int r = 0; r < 8; ++r) {
    lsm[wave][r + 8 * g][m]      = acc0[r];
    lsm[wave][r + 8 * g][16 + m] = acc1[r];
  }
  // combine half-wave partials: every lane ends with ||x||^2 of row m
  float x2 = x2p + __shfl_xor(x2p, 16, 32);
  __syncthreads();
  // softmax over k for row m (both halves compute, low half stores)
  float e[KCW];
  float mx = -3.4e38f;
#pragma unroll
  for (int k = 0; k < KCW; ++k) {
    float xc = lsm[wave][m][k];
    float lg = scale[k] * (x2 - 2.f * xc + c2[k]);
    e[k] = lg;
    mx = fmaxf(mx, lg);
  }
  float sum = 0.f;
#pragma unroll
  for (int k = 0; k < KCW; ++k) {
    float ex = __expf(e[k] - mx);
    e[k] = ex;
    sum += ex;
  }
  float inv = 1.f / sum;
  if (lane < 16) {
    float* Ar = A + ((size_t)b * NPOS + n0 + m) * KCW;
#pragma unroll
    for (int k = 0; k < KCW; ++k) Ar[k] = e[k] * inv;
  }
}

// ---------------------------------------------------------------- kernel 2
// out_raw[b][k][c] = sum_n A[b][n][k] * xf[b][n][c]  via f32 WMMA
__global__ __launch_bounds__(256) void enc_aggregate(
    const float* __restrict__ x, const float* __restrict__ A,
    float* __restrict__ out) {
  const int lane = threadIdx.x & 31;
  const int wave = threadIdx.x >> 5;
  const int wid  = blockIdx.x * 8 + wave;        // 256 waves total
  const int b    = wid >> 4;
  const int rem  = wid & 15;
  const int k0   = (rem >> 3) << 4;              // 0 or 16
  const int c0   = (rem & 7) << 6;               // 64-wide c block
  const int g    = lane >> 4;
  const int m    = lane & 15;
  const float* Ab = A + (size_t)b * NPOS * KCW + k0 + m;   // coalesced k runs
  const float* xb = x + (size_t)b * CHAN * NPOS + 2 * g;
  v8f acc[4] = {v8f{}, v8f{}, v8f{}, v8f{}};
  for (int n = 0; n < NPOS; n += 4) {
    if ((n & 255) == 0 && n + 256 < NPOS) {
      __builtin_prefetch(Ab + (size_t)(n + 256) * KCW, 0, 1);
      __builtin_prefetch(xb + (size_t)(c0 + m) * NPOS + n + 256, 0, 1);
    }
    // A-frag = A^T (K x n): element (row k=m, kdim j+2g) -> A[n+2g+j][k0+m]
    v2f af;
    af.x = Ab[(size_t)(n + 2 * g) * KCW];
    af.y = Ab[(size_t)(n + 2 * g + 1) * KCW];
#pragma unroll
    for (int t = 0; t < 4; ++t) {
      // B-frag = xf (n x C): element (kdim j+2g, col c) -> x[b][c][n+2g+j]
      const float* p = xb + (size_t)(c0 + t * 16 + m) * NPOS + n;
      v2f bf; bf.x = p[0]; bf.y = p[1];
      acc[t] = wmma_f32(af, bf, acc[t]);
    }
  }
  float* ob = out + ((size_t)b * KCW + k0) * CHAN + c0 + m;
#pragma unroll
  for (int t = 0; t < 4; ++t)
#pragma unroll
    for (int r = 0; r < 8; ++r)
      ob[(size_t)(r + 8 * g) * CHAN + t * 16] = acc[t][r];
}

// ---------------------------------------------------------------- kernel 3
// Asum[b][k] = sum_n A; out[b][k][c] -= Asum[b][k]*cw[k][c]
__global__ __launch_bounds__(256) void enc_finalize(
    const float* __restrict__ A, const float* __restrict__ cw,
    float* __restrict__ out) {
  __shared__ float part[8][KCW];
  __shared__ float asum[KCW];
  const int t = threadIdx.x;
  const int b = blockIdx.x;
  const int k = t & (KCW - 1);
  const int p = t >> 5;
  const float* Ab = A + (size_t)b * NPOS * KCW;
  float s = 0.f;
  for (int n = p; n < NPOS; n += 8) s += Ab[(size_t)n * KCW + k];
  part[p][k] = s;
  __syncthreads();
  if (t < KCW) {
    float tot = 0.f;
#pragma unroll
    for (int i = 0; i < 8; ++i) tot += part[i][t];
    asum[t] = tot;
  }
  __syncthreads();
  float* ob = out + (size_t)b * KCW * CHAN;
  for (int i = t; i < KCW * CHAN; i += 256) {
    int kk = i >> 9;                             // C = 512
    ob[i] -= asum[kk] * cw[i];
  }
}

// ---------------------------------------------------------------- launch
extern "C" void kernel_launch(void* const* d_in, const int* in_sizes, int n_in,
                              void* d_out, int out_size, void* d_ws,
                              size_t ws_size, hipStream_t stream) {
  const float* x     = (const float*)d_in[0];   // (B, C, H, W) f32
  const float* cw    = (const float*)d_in[1];   // (K, C) f32
  const float* scale = (const float*)d_in[2];   // (K,) f32
  float* out = (float*)d_out;                   // (B, K, C) f32
  float* wsf = (float*)d_ws;
  float* c2  = wsf;                             // 32 floats
  float* A   = wsf + 64;                        // B*N*K floats (~8 MB)

  enc_c2<<<1, 256, 0, stream>>>(cw, c2);
  enc_assign<<<(BATCH * NPOS) / 128, 256, 0, stream>>>(x, cw, scale, c2, A);
  enc_aggregate<<<32, 256, 0, stream>>>(x, A, out);
  enc_finalize<<<BATCH, 256, 0, stream>>>(A, cw, out);
}